// RectConv2d_67491116089421
// MI455X (gfx1250) — compile-verified
//
#include <hip/hip_runtime.h>
#include <hip/hip_bf16.h>
#include <math.h>

// ---------------------------------------------------------------------------
// MI455X / gfx1250, wave32. Matrix ops via V_WMMA_F32_16X16X4_F32 (full fp32:
// ~17 GFLOP total, L2-resident -> precision is free). K-order permuted so all
// fragment loads are aligned b64 with immediate offsets (no per-step VALU).
// Deform GEMM loop is software-pipelined (batched fragment loads) so WMMA
// issue overlaps LDS/global load latency instead of s_wait..0 per step.
// ---------------------------------------------------------------------------

typedef __attribute__((ext_vector_type(2))) float v2f;
typedef __attribute__((ext_vector_type(8))) float v8f;

#define ACT_NONE 0
#define ACT_RELU 1
#define ACT_SIN  2

#define B_  8
#define C_  64
#define HW_ 64
#define PIX_TOTAL (B_ * HW_ * HW_)   // 32768
#define KCONV 576                     // 64 ch * 9 taps
#define PI_F 3.1415926f

// LDS input tile: [3 halo rows][66 padded cols][64 ch + 4 pad]
#define XS_CSTRIDE 68
#define XS_ELEMS   (3 * 66 * XS_CSTRIDE)   // 13464 floats (~53 KB)

// ---------------------------------------------------------------------------
// Weight packers -> B-fragment-friendly layout: wB[(k>>2)*256 + o*4 + (k&3)]
// conv uses permuted k' = (di*3+dj)*64 + c ;  w3d uses k = c*N + n
// ---------------------------------------------------------------------------
__global__ void pack_conv_w(const float* __restrict__ w, float* __restrict__ wB) {
    int kp = blockIdx.x;                 // k' = off*64 + c
    int o  = threadIdx.x;                // 64 threads
    int off = kp >> 6, c = kp & 63;
    wB[(size_t)(kp >> 2) * 256 + o * 4 + (kp & 3)] = w[(size_t)o * KCONV + c * 9 + off];
}

__global__ void pack_w3d(const float* __restrict__ w, float* __restrict__ wB, int K) {
    int k = blockIdx.x;                  // k = c*N + n ; src flat = o*K + k
    int o = threadIdx.x;
    wB[(size_t)(k >> 2) * 256 + o * 4 + (k & 3)] = w[(size_t)o * K + k];
}

// ---------------------------------------------------------------------------
// Implicit-GEMM 3x3 conv 64->64, pad 1. One workgroup = one (batch,row):
// M = 64 pixels, N = 64 outs, K = 576 (permuted).  4 waves; wave = out-group.
// ---------------------------------------------------------------------------
template <int ACT>
__global__ __launch_bounds__(128) void conv3x3_wmma(
    const float* __restrict__ xin, const float* __restrict__ wB,
    const float* __restrict__ bias, float* __restrict__ y)
{
    __shared__ float xs[XS_ELEMS];
    const int row = blockIdx.x;
    const int b   = blockIdx.y;
    const int tid = threadIdx.x;

    // halo load: coalesced global reads (col fastest), channel-last LDS writes
    for (int idx = tid; idx < C_ * 3 * 66; idx += 128) {
        int col = idx % 66;
        int cr  = idx / 66;
        int r   = cr % 3;
        int c   = cr / 3;
        int gi  = row + r - 1;
        int gj  = col - 1;
        float v = 0.f;
        if (gi >= 0 && gi < HW_ && gj >= 0 && gj < HW_)
            v = xin[(((size_t)b * C_ + c) * HW_ + gi) * HW_ + gj];
        xs[(r * 66 + col) * XS_CSTRIDE + c] = v;
    }
    __syncthreads();

    const int wid   = tid >> 5;
    const int lane  = tid & 31;
    const int mrow  = lane & 15;
    const int khalf = (lane >> 4) << 1;          // 0 / 2
    const int o     = wid * 16 + mrow;

    v8f acc0 = {}, acc1 = {}, acc2 = {}, acc3 = {};
    const float* wBl = wB + o * 4 + khalf;       // + kgrp*256 per step

    int kgrp = 0;
    for (int off = 0; off < 9; ++off) {
        const int di = off / 3, dj = off % 3;
        const int abase = (di * 66 + dj + mrow) * XS_CSTRIDE + khalf;
        if (kgrp + 16 < 144)
            __builtin_prefetch(wBl + (size_t)(kgrp + 16) * 256, 0, 1);
#pragma unroll
        for (int cb = 0; cb < 64; cb += 4) {
            v2f bf = *(const v2f*)(wBl + (size_t)kgrp * 256);
            v2f a0 = *(const v2f*)(xs + abase + cb);
            v2f a1 = *(const v2f*)(xs + abase + 16 * XS_CSTRIDE + cb);
            v2f a2 = *(const v2f*)(xs + abase + 32 * XS_CSTRIDE + cb);
            v2f a3 = *(const v2f*)(xs + abase + 48 * XS_CSTRIDE + cb);
            acc0 = __builtin_amdgcn_wmma_f32_16x16x4_f32(false, a0, false, bf, (short)0, acc0, false, false);
            acc1 = __builtin_amdgcn_wmma_f32_16x16x4_f32(false, a1, false, bf, (short)0, acc1, false, false);
            acc2 = __builtin_amdgcn_wmma_f32_16x16x4_f32(false, a2, false, bf, (short)0, acc2, false, false);
            acc3 = __builtin_amdgcn_wmma_f32_16x16x4_f32(false, a3, false, bf, (short)0, acc3, false, false);
            ++kgrp;
        }
    }

    // epilogue: bias+act, stage D through LDS, coalesced b128 stores
    const float bo = bias[o];
    v8f accs[4] = {acc0, acc1, acc2, acc3};
    __syncthreads();                              // all A reads done before reuse
    float* sm = xs;                               // [o][68] (bank-spread, 16B-alignable)
#pragma unroll
    for (int pg = 0; pg < 4; ++pg)
#pragma unroll
        for (int r = 0; r < 8; ++r) {
            int m = r + ((lane >> 4) << 3);
            int p = pg * 16 + m;
            float v = accs[pg][r] + bo;
            if (ACT == ACT_RELU) v = fmaxf(v, 0.f);
            if (ACT == ACT_SIN)  v = __sinf(v);
            sm[o * 68 + p] = v;
        }
    __syncthreads();
    float* yrow = y + (((size_t)b * C_) * HW_ + row) * HW_;
    for (int idx = tid; idx < 64 * 16; idx += 128) {
        int o2 = idx >> 4, p4 = (idx & 15) << 2;
        float4 v = *(const float4*)(sm + o2 * 68 + p4);
        *(float4*)(yrow + (size_t)o2 * (HW_ * HW_) + p4) = v;
    }
}

// ---------------------------------------------------------------------------
// Heads: l / w / theta maps. One thread per pixel, 3 fused 64->1 convs.
// ---------------------------------------------------------------------------
__global__ __launch_bounds__(256) void heads_kernel(
    const float* __restrict__ off,
    const float* __restrict__ lw1, const float* __restrict__ lb1,
    const float* __restrict__ lw2, const float* __restrict__ lb2,
    const float* __restrict__ ww1, const float* __restrict__ wb1,
    const float* __restrict__ ww2, const float* __restrict__ wb2,
    const float* __restrict__ tw1, const float* __restrict__ tb1,
    const float* __restrict__ tw2, const float* __restrict__ tb2,
    float* __restrict__ lmap, float* __restrict__ wmap, float* __restrict__ tmap)
{
    int pix = blockIdx.x * blockDim.x + threadIdx.x;
    if (pix >= PIX_TOTAL) return;
    int b = pix >> 12, rem = pix & 4095, i = rem >> 6, j = rem & 63;

    float al = lb1[0], aw = wb1[0], at = tb1[0];
    for (int c = 0; c < C_; ++c) {
        const float* oc = off + (((size_t)b * C_ + c) * HW_) * HW_;
        for (int di = 0; di < 3; ++di) {
            int gi = i + di - 1;
            if (gi < 0 || gi >= HW_) continue;
            for (int dj = 0; dj < 3; ++dj) {
                int gj = j + dj - 1;
                if (gj < 0 || gj >= HW_) continue;
                float v = oc[gi * HW_ + gj];
                int kidx = c * 9 + di * 3 + dj;
                al = fmaf(v, lw1[kidx], al);
                aw = fmaf(v, ww1[kidx], aw);
                at = fmaf(v, tw1[kidx], at);
            }
        }
    }
    float hl = __sinf(al) * lw2[0] + lb2[0];
    float hw = __sinf(aw) * ww2[0] + wb2[0];
    float ht = __sinf(at) * tw2[0] + tb2[0];
    lmap[pix] = (1.f / (1.f + __expf(-hl))) * 12.f + 2.f;   // sigmoid*(LMAX-2)+2
    wmap[pix] = (1.f / (1.f + __expf(-hw))) * 12.f + 2.f;
    tmap[pix] = (ht / (1.f + fabsf(ht))) * PI_F;            // softsign * pi
}

// ---------------------------------------------------------------------------
// Deterministic single-block reduction -> NX, NY (pick_N semantics)
// ---------------------------------------------------------------------------
__global__ __launch_bounds__(256) void stats_kernel(
    const float* __restrict__ lmap, const float* __restrict__ wmap, int* __restrict__ stats)
{
    __shared__ float sl[256], sw[256];
    int tid = threadIdx.x;
    float a = 0.f, c = 0.f;
    for (int idx = tid; idx < PIX_TOTAL; idx += 256) { a += lmap[idx]; c += wmap[idx]; }
    sl[tid] = a; sw[tid] = c;
    __syncthreads();
    for (int s = 128; s > 0; s >>= 1) {
        if (tid < s) { sl[tid] += sl[tid + s]; sw[tid] += sw[tid + s]; }
        __syncthreads();
    }
    if (tid == 0) {
        float ml = sl[0] * (1.f / PIX_TOTAL);
        float mw = sw[0] * (1.f / PIX_TOTAL);
        stats[0] = (ml < 6.f) ? 3 : (int)floorf(ml * 0.5f);
        stats[1] = (mw < 6.f) ? 3 : (int)floorf(mw * 0.5f);
    }
}

// ---------------------------------------------------------------------------
// Deform sampling + 'bchwn,ocn->bohw' contraction (WMMA) + fused *m + bias.
// 16 pixels/block, 4 waves. A-tile stride K+2 (even -> b64 legal, banks spread).
// K-loop unrolled x4 with batched fragment loads (software pipelining).
// ---------------------------------------------------------------------------
__global__ __launch_bounds__(128) void deform_wmma(
    const float* __restrict__ x, const float* __restrict__ wB3d,
    const float* __restrict__ lmap, const float* __restrict__ wmap,
    const float* __restrict__ tmap, const float* __restrict__ mbuf,
    const float* __restrict__ bbuf, const int* __restrict__ stats,
    int N, float* __restrict__ out)
{
    extern __shared__ float a_lds[];             // 16 * (K+2), reused as [64][20] for stores
    const int K  = C_ * N;
    const int AS = K + 2;
    const int NX = stats[0];
    const int NY = stats[1];
    const int tid  = threadIdx.x;
    const int tile = blockIdx.x;

    for (int item = tid; item < 16 * N; item += blockDim.x) {
        int m = item / N, n = item % N;
        int pix = tile * 16 + m;
        int b = pix >> 12, rem = pix & 4095, i = rem >> 6, j = rem & 63;
        float lv = lmap[pix], wv = wmap[pix], th = tmap[pix];
        int ix = n / NY, iy = n % NY;
        float pnx = (float)(ix - (NX >> 1));     // arange(-(NX-1)//2, ...) start
        float pny = (float)(iy - (NY >> 1));
        float p_x = (float)(i + 1) + (lv / (float)NX) * pnx;
        float p_y = (float)(j + 1) + (wv / (float)NY) * pny;
        float ct = __cosf(th), st = __sinf(th);
        float px = p_x * ct - p_y * st;
        float py = p_x * st + p_y * ct;
        float fx = floorf(px), fy = floorf(py);
        float ltx = fminf(fmaxf(fx, 0.f), 65.f);
        float lty = fminf(fmaxf(fy, 0.f), 65.f);
        float rbx = fminf(fmaxf(fx + 1.f, 0.f), 65.f);
        float rby = fminf(fmaxf(fy + 1.f, 0.f), 65.f);
        float pxc = fminf(fmaxf(px, 0.f), 65.f);
        float pyc = fminf(fmaxf(py, 0.f), 65.f);
        float g_lt = (1.f + (ltx - pxc)) * (1.f + (lty - pyc));
        float g_rb = (1.f - (rbx - pxc)) * (1.f - (rby - pyc));
        float g_lb = (1.f + (ltx - pxc)) * (1.f - (rby - pyc));
        float g_rt = (1.f - (rbx - pxc)) * (1.f + (lty - pyc));
        int qx0 = (int)ltx, qy0 = (int)lty, qx1 = (int)rbx, qy1 = (int)rby;
        bool i00 = (qx0 >= 1) & (qx0 <= 64) & (qy0 >= 1) & (qy0 <= 64);
        bool i11 = (qx1 >= 1) & (qx1 <= 64) & (qy1 >= 1) & (qy1 <= 64);
        bool i01 = (qx0 >= 1) & (qx0 <= 64) & (qy1 >= 1) & (qy1 <= 64);
        bool i10 = (qx1 >= 1) & (qx1 <= 64) & (qy0 >= 1) & (qy0 <= 64);
        int o00 = (qx0 - 1) * HW_ + (qy0 - 1);
        int o11 = (qx1 - 1) * HW_ + (qy1 - 1);
        int o01 = (qx0 - 1) * HW_ + (qy1 - 1);
        int o10 = (qx1 - 1) * HW_ + (qy0 - 1);
        const float* xb = x + (size_t)b * C_ * HW_ * HW_;
        for (int c = 0; c < C_; ++c) {
            const float* xc = xb + (size_t)c * HW_ * HW_;
            float v00 = i00 ? xc[o00] : 0.f;
            float v11 = i11 ? xc[o11] : 0.f;
            float v01 = i01 ? xc[o01] : 0.f;
            float v10 = i10 ? xc[o10] : 0.f;
            a_lds[m * AS + c * N + n] =
                g_lt * v00 + g_rb * v11 + g_lb * v01 + g_rt * v10;
        }
    }
    __syncthreads();

    const int wid   = tid >> 5;
    const int lane  = tid & 31;
    const int mrow  = lane & 15;
    const int khalf = (lane >> 4) << 1;
    const int o     = wid * 16 + mrow;

    v8f acc = {};
    const float* arow = a_lds + mrow * AS + khalf;
    const float* wBl  = wB3d + o * 4 + khalf;
    // K = 64*N -> always a multiple of 16 k (4 wmma steps); batch loads so
    // WMMA issue overlaps load latency instead of s_wait..0 per step.
    for (int kk = 0; kk < K; kk += 16) {
        v2f af[4], bf[4];
#pragma unroll
        for (int u = 0; u < 4; ++u) {
            af[u] = *(const v2f*)(arow + kk + 4 * u);
            bf[u] = *(const v2f*)(wBl + (size_t)((kk >> 2) + u) * 256);
        }
#pragma unroll
        for (int u = 0; u < 4; ++u)
            acc = __builtin_amdgcn_wmma_f32_16x16x4_f32(false, af[u], false, bf[u], (short)0, acc, false, false);
    }

    // stage D through LDS, then fused (*m + bias) with coalesced b128 traffic
    __syncthreads();
    float* sm = a_lds;                           // [o][20]
#pragma unroll
    for (int r = 0; r < 8; ++r) {
        int m = r + ((lane >> 4) << 3);
        sm[o * 20 + m] = acc[r];
    }
    __syncthreads();
    int pix0 = tile * 16;
    int b = pix0 >> 12, i = (pix0 & 4095) >> 6, j0 = pix0 & 63;
    size_t rowbase = (((size_t)b * C_) * HW_ + i) * HW_ + j0;
    for (int idx = tid; idx < 64 * 4; idx += 128) {
        int o2 = idx >> 2, m4 = (idx & 3) << 2;
        float4 a = *(const float4*)(sm + o2 * 20 + m4);
        size_t g = rowbase + (size_t)o2 * (HW_ * HW_) + m4;
        float4 mm = *(const float4*)(mbuf + g);
        float4 bb = *(const float4*)(bbuf + g);
        float4 r4;
        r4.x = a.x * mm.x + bb.x;
        r4.y = a.y * mm.y + bb.y;
        r4.z = a.z * mm.z + bb.z;
        r4.w = a.w * mm.w + bb.w;
        *(float4*)(out + g) = r4;
    }
}

// ---------------------------------------------------------------------------
// Host orchestration
// ---------------------------------------------------------------------------
extern "C" void kernel_launch(void* const* d_in, const int* in_sizes, int n_in,
                              void* d_out, int out_size, void* d_ws, size_t ws_size,
                              hipStream_t stream) {
    const float* x = (const float*)d_in[0];
    const float* m_w1 = (const float*)d_in[1];  const float* m_b1 = (const float*)d_in[2];
    const float* m_w2 = (const float*)d_in[3];  const float* m_b2 = (const float*)d_in[4];
    const float* b_w1 = (const float*)d_in[5];  const float* b_b1 = (const float*)d_in[6];
    const float* b_w2 = (const float*)d_in[7];  const float* b_b2 = (const float*)d_in[8];
    const float* p_w1 = (const float*)d_in[9];  const float* p_b1 = (const float*)d_in[10];
    const float* p_w2 = (const float*)d_in[11]; const float* p_b2 = (const float*)d_in[12];
    const float* l_w1 = (const float*)d_in[13]; const float* l_b1 = (const float*)d_in[14];
    const float* l_w2 = (const float*)d_in[15]; const float* l_b2 = (const float*)d_in[16];
    const float* w_w1 = (const float*)d_in[17]; const float* w_b1 = (const float*)d_in[18];
    const float* w_w2 = (const float*)d_in[19]; const float* w_b2 = (const float*)d_in[20];
    const float* t_w1 = (const float*)d_in[21]; const float* t_b1 = (const float*)d_in[22];
    const float* t_w2 = (const float*)d_in[23]; const float* t_b2 = (const float*)d_in[24];
    const float* w3d  = (const float*)d_in[25];
    float* out = (float*)d_out;

    const int N   = in_sizes[25] / (C_ * C_);    // NX*NY from w3d shape
    const int K3d = C_ * N;

    const size_t F = (size_t)B_ * C_ * HW_ * HW_;
    float* ws_h    = (float*)d_ws;
    float* ws_m    = ws_h + F;
    float* ws_bias = ws_m + F;
    float* ws_off  = ws_bias + F;
    float* ws_l    = ws_off + F;
    float* ws_w    = ws_l + PIX_TOTAL;
    float* ws_t    = ws_w + PIX_TOTAL;
    float* ws_wB   = ws_t + PIX_TOTAL;
    size_t wB_elems = (size_t)64 * (size_t)((KCONV > K3d) ? KCONV : K3d);
    int*   ws_stats = (int*)(ws_wB + wB_elems);

    dim3 cgrid(HW_, B_);

    // m branch
    pack_conv_w<<<KCONV, 64, 0, stream>>>(m_w1, ws_wB);
    conv3x3_wmma<ACT_RELU><<<cgrid, 128, 0, stream>>>(x, ws_wB, m_b1, ws_h);
    pack_conv_w<<<KCONV, 64, 0, stream>>>(m_w2, ws_wB);
    conv3x3_wmma<ACT_NONE><<<cgrid, 128, 0, stream>>>(ws_h, ws_wB, m_b2, ws_m);

    // b branch
    pack_conv_w<<<KCONV, 64, 0, stream>>>(b_w1, ws_wB);
    conv3x3_wmma<ACT_RELU><<<cgrid, 128, 0, stream>>>(x, ws_wB, b_b1, ws_h);
    pack_conv_w<<<KCONV, 64, 0, stream>>>(b_w2, ws_wB);
    conv3x3_wmma<ACT_NONE><<<cgrid, 128, 0, stream>>>(ws_h, ws_wB, b_b2, ws_bias);

    // p branch
    pack_conv_w<<<KCONV, 64, 0, stream>>>(p_w1, ws_wB);
    conv3x3_wmma<ACT_SIN><<<cgrid, 128, 0, stream>>>(x, ws_wB, p_b1, ws_h);
    pack_conv_w<<<KCONV, 64, 0, stream>>>(p_w2, ws_wB);
    conv3x3_wmma<ACT_NONE><<<cgrid, 128, 0, stream>>>(ws_h, ws_wB, p_b2, ws_off);

    // heads + pick_N
    heads_kernel<<<PIX_TOTAL / 256, 256, 0, stream>>>(
        ws_off, l_w1, l_b1, l_w2, l_b2, w_w1, w_b1, w_w2, w_b2,
        t_w1, t_b1, t_w2, t_b2, ws_l, ws_w, ws_t);
    stats_kernel<<<1, 256, 0, stream>>>(ws_l, ws_w, ws_stats);

    // w3d pack + deform/contract/epilogue
    pack_w3d<<<K3d, 64, 0, stream>>>(w3d, ws_wB, K3d);
    size_t lds_floats = (size_t)16 * (K3d + 2);
    if (lds_floats < 64 * 20) lds_floats = 64 * 20;     // store-staging reuse
    deform_wmma<<<PIX_TOTAL / 16, 128, lds_floats * sizeof(float), stream>>>(
        x, ws_wB, ws_l, ws_w, ws_t, ws_m, ws_bias, ws_stats, N, out);
}